// GCN_LPA_21053929685140
// MI455X (gfx1250) — compile-verified
//
#include <hip/hip_runtime.h>
#include <stdint.h>

// ---------------------------------------------------------------------------
// GCN (2x GCNConv with symmetric norm + self loops) + edge-softmax LPA (5 it)
// MI455X / gfx1250: wave32, WMMA f32 16x16x4 for the dense GEMMs.
// Round 2: LDS A-tile padded (bank-conflict-free ds_load_2addr_b64),
//          uniform-branch full-tile store path (no per-element exec dance).
// ---------------------------------------------------------------------------

typedef float v2f __attribute__((ext_vector_type(2)));
typedef float v8f __attribute__((ext_vector_type(8)));

#define THREADS 256

static __device__ __forceinline__ long long gtid() {
  return (long long)blockIdx.x * blockDim.x + threadIdx.x;
}

// ------------------------------ utility ------------------------------------
__global__ void k_fill(float* __restrict__ p, float v, long long n) {
  long long i = gtid();
  if (i < n) p[i] = v;
}

__global__ void k_deg_scatter(const long long* __restrict__ dst,
                              const float* __restrict__ ew,
                              float* __restrict__ deg, long long E) {
  long long e = gtid();
  if (e < E) atomicAdd(&deg[dst[e]], ew[e]);
}

__global__ void k_dinv(const float* __restrict__ deg, float* __restrict__ dinv,
                       long long n) {
  long long i = gtid();
  if (i < n) {
    float d = deg[i];
    dinv[i] = (d > 0.0f) ? rsqrtf(fmaxf(d, 1e-12f)) : 0.0f;
  }
}

__global__ void k_norm(const long long* __restrict__ src,
                       const long long* __restrict__ dst,
                       const float* __restrict__ ew,
                       const float* __restrict__ dinv,
                       float* __restrict__ nw, long long E) {
  long long e = gtid();
  if (e < E) nw[e] = dinv[src[e]] * ew[e] * dinv[dst[e]];
}

// ------------------------------ WMMA GEMM ----------------------------------
// Out[nrows x F] = X[nrows x K] @ W[K x F], row-major, fp32.
// One 16-row M-tile per block; each wave owns one 16-col N-tile.
// A tile staged in LDS (coalesced), K stepped by 4 via v_wmma_f32_16x16x4_f32.
// Fragment layouts per CDNA5 ISA 7.12.2:
//   A 16x4 : lane l -> row M=l&15, holds K = (l>>4)*2 + {0,1} in {v0,v1}
//   B 4x16 : lane l -> col N=l&15, holds K = (l>>4)*2 + {0,1} in {v0,v1}
//   C/D    : vgpr r -> M = r + (l>>4)*8, N = l&15
// LDS row stride padded to K+4 floats: (K+4) % 64 == 4, so the 16 lanes of a
// half-wave land on banks {4l,4l+1} (koff=2 half: {4l+2,4l+3}) -> conflict-free.
template <int K, int F>
__global__ __launch_bounds__((F / 16) * 32) void k_gemm_wmma(
    const float* __restrict__ X, const float* __restrict__ W,
    float* __restrict__ Out, int nrows) {
  constexpr int LDK = K + 4;
  __shared__ float tileA[16 * LDK];
  const int m0 = blockIdx.x << 4;
  const int tid = threadIdx.x;
  constexpr int NTHR = (F / 16) * 32;

  const bool fullTile = (m0 + 16 <= nrows);  // uniform per block
  if (fullTile) {
    for (int i = tid; i < 16 * K; i += NTHR) {
      int r = i / K, k = i - r * K;
      tileA[r * LDK + k] = X[(size_t)(m0 + r) * K + k];
    }
  } else {
    for (int i = tid; i < 16 * K; i += NTHR) {
      int r = i / K, k = i - r * K;
      int row = m0 + r;
      tileA[r * LDK + k] = (row < nrows) ? X[(size_t)row * K + k] : 0.0f;
    }
  }
  __syncthreads();

  const int lane = tid & 31;
  const int wave = tid >> 5;
  const int mloc = lane & 15;
  const int koff = (lane >> 4) << 1;  // 0 or 2
  const int col = (wave << 4) + mloc;

  v8f acc = {};
  const float* Arow = &tileA[mloc * LDK + koff];
  const float* Bcol = W + col + (size_t)koff * F;
#pragma unroll 4
  for (int k = 0; k < K; k += 4) {
    v2f a, b;
    a.x = Arow[k];
    a.y = Arow[k + 1];
    b.x = Bcol[(size_t)k * F];
    b.y = Bcol[(size_t)(k + 1) * F];
    acc = __builtin_amdgcn_wmma_f32_16x16x4_f32(false, a, false, b, (short)0,
                                                acc, false, false);
  }

  const int mhi = (lane >> 4) << 3;
  float* Orow = Out + (size_t)(m0 + mhi) * F + col;
  if (fullTile) {  // uniform branch: straight-line 8-store clause
#pragma unroll
    for (int r = 0; r < 8; ++r) Orow[(size_t)r * F] = acc[r];
  } else {
#pragma unroll
    for (int r = 0; r < 8; ++r) {
      if (m0 + mhi + r < nrows) Orow[(size_t)r * F] = acc[r];
    }
  }
}

// ------------------------------ edge scatter -------------------------------
// Out[dst, :] += coef[e] * Hin[src, :]; one thread per (edge, 4-feature chunk).
template <int F>
__global__ void k_scatter_edges(const long long* __restrict__ src,
                                const long long* __restrict__ dst,
                                const float* __restrict__ coef,
                                const float* __restrict__ Hin,
                                float* __restrict__ Out, long long E) {
  constexpr int CH = F / 4;
  long long t = gtid();
  if (t >= E * CH) return;
  long long e = t / CH;
  int c = (int)(t - e * CH) << 2;
  long long s = src[e], d = dst[e];
  float w = coef[e];
  const float4 h = *(const float4*)(Hin + s * (long long)F + c);
  float* o = Out + d * (long long)F + c;
  atomicAdd(o + 0, w * h.x);
  atomicAdd(o + 1, w * h.y);
  atomicAdd(o + 2, w * h.z);
  atomicAdd(o + 3, w * h.w);
}

// A[i,f] = (relu?)(A[i,f] + dinv[i]^2 * H[i,f] + bias[f])  (self-loop + bias)
template <bool RELU>
__global__ void k_finalize(float* __restrict__ A, const float* __restrict__ H,
                           const float* __restrict__ dinv,
                           const float* __restrict__ bias, long long n, int F) {
  long long idx = gtid();
  if (idx >= n * F) return;
  long long i = idx / F;
  int f = (int)(idx - i * F);
  float di = dinv[i];
  float v = A[idx] + di * di * H[idx] + bias[f];
  if (RELU) v = fmaxf(v, 0.0f);
  A[idx] = v;
}

// ------------------------------ LPA ----------------------------------------
__device__ __forceinline__ void atomicMaxF(float* addr, float val) {
  int old = __float_as_int(*addr);
  while (__int_as_float(old) < val) {
    int assumed = old;
    old = atomicCAS((int*)addr, assumed, __float_as_int(val));
    if (old == assumed) break;
  }
}

__global__ void k_segmax(const long long* __restrict__ dst,
                         const float* __restrict__ ew, float* __restrict__ m,
                         long long E) {
  long long e = gtid();
  if (e < E) atomicMaxF(&m[dst[e]], ew[e]);
}

__global__ void k_expsum(const long long* __restrict__ dst,
                         const float* __restrict__ ew,
                         const float* __restrict__ m, float* __restrict__ ea,
                         float* __restrict__ s, long long E) {
  long long e = gtid();
  if (e < E) {
    long long d = dst[e];
    float v = __expf(ew[e] - m[d]);
    ea[e] = v;
    atomicAdd(&s[d], v);
  }
}

__global__ void k_alpha(const long long* __restrict__ dst,
                        const float* __restrict__ s, float* __restrict__ ea,
                        long long E) {
  long long e = gtid();
  if (e < E) ea[e] = ea[e] / (s[dst[e]] + 1e-16f);
}

__global__ void k_onehot(const long long* __restrict__ y,
                         const unsigned char* __restrict__ mask,
                         float* __restrict__ out, long long n, int C) {
  long long idx = gtid();
  if (idx >= n * C) return;
  long long i = idx / C;
  int c = (int)(idx - i * C);
  out[idx] = (mask[i] && (int)y[i] == c) ? 1.0f : 0.0f;
}

// ------------------------------ launch -------------------------------------
extern "C" void kernel_launch(void* const* d_in, const int* in_sizes, int n_in,
                              void* d_out, int out_size, void* d_ws,
                              size_t ws_size, hipStream_t stream) {
  (void)n_in; (void)out_size; (void)ws_size;

  const float* x = (const float*)d_in[0];
  const long long* ei = (const long long*)d_in[1];   // int64 [2, E]
  const long long* y = (const long long*)d_in[2];    // int64 [N]
  const unsigned char* mask = (const unsigned char*)d_in[3];  // bool [N]
  const float* ew = (const float*)d_in[4];
  const float* W1 = (const float*)d_in[5];
  const float* b1 = (const float*)d_in[6];
  const float* W2 = (const float*)d_in[7];
  const float* b2 = (const float*)d_in[8];

  const long long n = in_sizes[2];   // N (y length)
  const long long E = in_sizes[4];   // edge_weight length
  constexpr int INF = 512, HID = 256, C = 64, ITERS = 5;

  const long long* srcI = ei;
  const long long* dstI = ei + E;

  // workspace layout (floats); peak ~220 MB
  float* ws = (float*)d_ws;
  float* h1 = ws;                    // n*HID   (later reused for h2: n*C)
  float* a1 = h1 + n * HID;          // n*HID   (later reused for LPA ping/pong)
  float* deg = a1 + n * HID;         // n
  float* dinv = deg + n;             // n
  float* mseg = dinv + n;            // n
  float* sseg = mseg + n;            // n
  float* nw = sseg + n;              // E  (GCN norm; later reused for alpha)

  float* outH = (float*)d_out;       // n*C
  float* outY = outH + n * C;        // n*C

  auto nb = [](long long t) {
    return (unsigned int)((t + THREADS - 1) / THREADS);
  };

  // --- symmetric GCN normalization (with self loops, weight 1) ---
  k_fill<<<nb(n), THREADS, 0, stream>>>(deg, 1.0f, n);
  k_deg_scatter<<<nb(E), THREADS, 0, stream>>>(dstI, ew, deg, E);
  k_dinv<<<nb(n), THREADS, 0, stream>>>(deg, dinv, n);
  k_norm<<<nb(E), THREADS, 0, stream>>>(srcI, dstI, ew, dinv, nw, E);

  dim3 gTiles((unsigned int)((n + 15) / 16));

  // --- layer 1: h1 = x@W1 ; aggregate ; +b1 ; relu ---
  k_gemm_wmma<INF, HID><<<gTiles, (HID / 16) * 32, 0, stream>>>(x, W1, h1, (int)n);
  k_fill<<<nb(n * HID), THREADS, 0, stream>>>(a1, 0.0f, n * HID);
  k_scatter_edges<HID><<<nb(E * (HID / 4)), THREADS, 0, stream>>>(srcI, dstI, nw, h1, a1, E);
  k_finalize<true><<<nb(n * HID), THREADS, 0, stream>>>(a1, h1, dinv, b1, n, HID);

  // --- layer 2: h2 = a1@W2 ; aggregate into d_out ; +b2 ---
  float* h2 = h1;  // h1 region free now
  k_gemm_wmma<HID, C><<<gTiles, (C / 16) * 32, 0, stream>>>(a1, W2, h2, (int)n);
  k_fill<<<nb(n * C), THREADS, 0, stream>>>(outH, 0.0f, n * C);
  k_scatter_edges<C><<<nb(E * (C / 4)), THREADS, 0, stream>>>(srcI, dstI, nw, h2, outH, E);
  k_finalize<false><<<nb(n * C), THREADS, 0, stream>>>(outH, h2, dinv, b2, n, C);

  // --- LPA: per-dst softmax of edge weights, 5 propagation iterations ---
  float* alpha = nw;  // norm no longer needed
  k_fill<<<nb(n), THREADS, 0, stream>>>(mseg, -__builtin_inff(), n);
  k_segmax<<<nb(E), THREADS, 0, stream>>>(dstI, ew, mseg, E);
  k_fill<<<nb(n), THREADS, 0, stream>>>(sseg, 0.0f, n);
  k_expsum<<<nb(E), THREADS, 0, stream>>>(dstI, ew, mseg, alpha, sseg, E);
  k_alpha<<<nb(E), THREADS, 0, stream>>>(dstI, sseg, alpha, E);

  float* lpaA = a1;            // a1 region free now
  float* lpaB = a1 + n * C;
  k_onehot<<<nb(n * C), THREADS, 0, stream>>>(y, mask, lpaA, n, C);

  float* cur = lpaA;
  float* nxt = lpaB;
  for (int t = 0; t < ITERS; ++t) {
    float* target = (t == ITERS - 1) ? outY : nxt;
    k_fill<<<nb(n * C), THREADS, 0, stream>>>(target, 0.0f, n * C);
    k_scatter_edges<C><<<nb(E * (C / 4)), THREADS, 0, stream>>>(srcI, dstI, alpha, cur, target, E);
    nxt = cur;
    cur = target;
  }
}